// GCNLayer_13984413516308
// MI455X (gfx1250) — compile-verified
//
#include <hip/hip_runtime.h>
#include <hip/hip_bf16.h>

#define IN_DIM  128
#define OUT_DIM 128

typedef float v2f __attribute__((ext_vector_type(2)));
typedef float v8f __attribute__((ext_vector_type(8)));

// ---------------------------------------------------------------------------
// Kernel 1: support = x @ W   (M=50000, K=128, N=128), fp32 WMMA 16x16x4.
//
// Block = 256 threads (8 wave32), covering 128 rows x 128 cols.
// Wave w owns the 16-row stripe m0 = blockRow + 16w across ALL 128 columns,
// holding 8 independent 16x16 f32 accumulators (64 VGPRs). Per K-step the
// wave loads ONE A-fragment (global_load_b64, no cross-wave redundancy) and
// issues 8 independent v_wmma_f32_16x16x4_f32 — independent D registers, so
// the XDL pipe is kept full without RAW stalls (5-slot hazard window).
//
// W (64 KB fp32) is staged once in LDS; B-fragments come from ds loads.
//
// Fragment layouts per CDNA5 ISA §7.12.2 (wave32):
//   A 16x4 f32 : lanes 0-15 K={k0,k0+1}, lanes 16-31 K={k0+2,k0+3} (2 VGPRs)
//   B 4x16 f32 : mirrored, column = lane&15
//   C/D 16x16  : VGPR r -> row m0 + r (+8 for lanes 16-31), col = lane&15
// ---------------------------------------------------------------------------
__global__ void __launch_bounds__(256)
gcn_gemm_wmma(const float* __restrict__ x, const float* __restrict__ w,
              float* __restrict__ support, int n_nodes) {
  __shared__ float wlds[IN_DIM * OUT_DIM];   // 64 KB of 320 KB WGP LDS
  for (int i = threadIdx.x; i < IN_DIM * OUT_DIM; i += 256)
    wlds[i] = w[i];
  __syncthreads();

  const int lane = threadIdx.x & 31;
  const int wave = threadIdx.x >> 5;
  const int m0   = blockIdx.x * 128 + wave * 16;

  const int koff = (lane >> 4) * 2;          // 0 for lanes 0-15, 2 for 16-31
  const int nsub = lane & 15;                // column within each 16-wide tile

  int mrow = m0 + (lane & 15);
  if (mrow >= n_nodes) mrow = n_nodes - 1;   // clamp loads only; EXEC stays all-1s

  const float* __restrict__ xrow = x + (size_t)mrow * IN_DIM;

  v8f acc[8];
  #pragma unroll
  for (int t = 0; t < 8; ++t) acc[t] = (v8f){};

  #pragma unroll 2
  for (int k0 = 0; k0 < IN_DIM; k0 += 4) {
    v2f a;
    a.x = xrow[k0 + koff];                   // consecutive -> one global_load_b64
    a.y = xrow[k0 + koff + 1];
    const float* __restrict__ wk0 = &wlds[(k0 + koff) * OUT_DIM + nsub];
    #pragma unroll
    for (int t = 0; t < 8; ++t) {            // 8 independent WMMA chains
      v2f b;
      b.x = wk0[t * 16];
      b.y = wk0[t * 16 + OUT_DIM];
      acc[t] = __builtin_amdgcn_wmma_f32_16x16x4_f32(
          /*neg_a=*/false, a, /*neg_b=*/false, b,
          /*c_mod=*/(short)0, acc[t], /*reuse_a=*/false, /*reuse_b=*/false);
    }
  }

  const int mbase = m0 + ((lane >> 4) << 3); // +8 row offset for upper half-wave
  float* outp = support + (size_t)mbase * OUT_DIM + nsub;

  if (m0 + 15 < n_nodes) {
    // Fast path (all blocks except the last partial one): straight stores.
    #pragma unroll
    for (int r = 0; r < 8; ++r)
      #pragma unroll
      for (int t = 0; t < 8; ++t)
        outp[(size_t)r * OUT_DIM + t * 16] = acc[t][r];
  } else {
    #pragma unroll
    for (int r = 0; r < 8; ++r)
      if (mbase + r < n_nodes)
        #pragma unroll
        for (int t = 0; t < 8; ++t)
          outp[(size_t)r * OUT_DIM + t * 16] = acc[t][r];
  }
}

// ---------------------------------------------------------------------------
// Kernel 2: out[i, :] = bias[:]  (zero-init of poisoned d_out + bias term)
// ---------------------------------------------------------------------------
__global__ void __launch_bounds__(256)
gcn_bias_init(const float* __restrict__ bias, float* __restrict__ out, int total) {
  int idx = blockIdx.x * 256 + threadIdx.x;
  if (idx < total)
    out[idx] = bias[idx & (OUT_DIM - 1)];
}

// ---------------------------------------------------------------------------
// Kernel 3: COO SpMM scatter. One wave per edge: 32 lanes x float4 = 128 dims.
// support + out are L2-resident (25.6 MB each vs 192 MB L2), so the gather
// (global_load_b128) and the no-return fp32 atomic adds are L2-served.
// ---------------------------------------------------------------------------
__global__ void __launch_bounds__(256)
gcn_spmm_scatter(const float* __restrict__ support,
                 const float* __restrict__ adj_vals,
                 const int* __restrict__ edge_row,
                 const int* __restrict__ edge_col,
                 float* __restrict__ out, int n_edges) {
  const int lane = threadIdx.x & 31;
  const int edge = blockIdx.x * 8 + (threadIdx.x >> 5);
  if (edge >= n_edges) return;

  const int   col = edge_col[edge];
  const int   row = edge_row[edge];
  const float v   = adj_vals[edge];

  const float4 m =
      *((const float4*)(support + (size_t)col * OUT_DIM) + lane);

  float* dst = out + (size_t)row * OUT_DIM + lane * 4;
  // Native global_atomic_add_f32, no return (relaxed, device scope).
  __hip_atomic_fetch_add(dst + 0, m.x * v, __ATOMIC_RELAXED, __HIP_MEMORY_SCOPE_AGENT);
  __hip_atomic_fetch_add(dst + 1, m.y * v, __ATOMIC_RELAXED, __HIP_MEMORY_SCOPE_AGENT);
  __hip_atomic_fetch_add(dst + 2, m.z * v, __ATOMIC_RELAXED, __HIP_MEMORY_SCOPE_AGENT);
  __hip_atomic_fetch_add(dst + 3, m.w * v, __ATOMIC_RELAXED, __HIP_MEMORY_SCOPE_AGENT);
}

// ---------------------------------------------------------------------------
// Launch
// ---------------------------------------------------------------------------
extern "C" void kernel_launch(void* const* d_in, const int* in_sizes, int n_in,
                              void* d_out, int out_size, void* d_ws, size_t ws_size,
                              hipStream_t stream) {
  const float* x        = (const float*)d_in[0];
  const float* w        = (const float*)d_in[1];
  const float* bias     = (const float*)d_in[2];
  const float* adj_vals = (const float*)d_in[3];
  const int*   edge_row = (const int*)d_in[4];
  const int*   edge_col = (const int*)d_in[5];

  const int n_nodes = in_sizes[0] / IN_DIM;
  const int n_edges = in_sizes[3];

  float* out     = (float*)d_out;
  float* support = (float*)d_ws;   // n_nodes * OUT_DIM floats = 25.6 MB

  // 1) support = x @ W  (WMMA fp32, 128-row stripes per block)
  const int nblocks = (n_nodes + 127) / 128;
  gcn_gemm_wmma<<<nblocks, 256, 0, stream>>>(x, w, support, n_nodes);

  // 2) out = bias (broadcast)  -- also clears poisoned d_out
  const int total = n_nodes * OUT_DIM;
  gcn_bias_init<<<(total + 255) / 256, 256, 0, stream>>>(bias, out, total);

  // 3) out[row] += adj * support[col]  (wave-per-edge atomic scatter)
  gcn_spmm_scatter<<<(n_edges + 7) / 8, 256, 0, stream>>>(
      support, adj_vals, edge_row, edge_col, out, n_edges);
}